// TimeFrquencyFusion_33775622816160
// MI455X (gfx1250) — compile-verified
//
#include <hip/hip_runtime.h>
#include <hip/hip_bf16.h>

// Elementwise time/frequency fusion: out = 0.5*(softmax-blend) + 0.5*(sigmoid-gated blend)
// Pure HBM-streaming kernel: 238 MB traffic @ 23.3 TB/s ~ 10 us floor, ~0.2 GFLOP compute.
// CDNA5 paths: b128 NT loads/stores (bypass 192MB L2 for single-use lines),
// global_prefetch_b8 one unrolled iteration ahead, scalar-pipe uniform parameter math
// (s_load + v_s_exp_f32), v_exp_f32/v_rcp_f32 trans sigmoid, 2x unroll for 4
// in-flight b128 loads per wave (partial s_wait_loadcnt overlap).

typedef float v4f __attribute__((ext_vector_type(4)));

__device__ __forceinline__ v4f fuse4(v4f tv, v4f fv,
                                     float g0, float g1, float gbv,
                                     float c_t, float c_f)
{
    v4f ov;
#pragma unroll
    for (int k = 0; k < 4; ++k) {
        const float t = tv[k];
        const float f = fv[k];
        // gate = sigmoid(t*g0 + f*g1 + gb)   (v_exp_f32 + v_rcp_f32, trans unit)
        const float z    = fmaf(t, g0, fmaf(f, g1, gbv));
        const float gate = __builtin_amdgcn_rcpf(1.0f + __expf(-z));
        // out = t*(0.5*w0 + 0.5*gate) + f*(0.5*w1 + 0.5*(1-gate))
        const float ct = fmaf(0.5f, gate, c_t);
        const float cf = fmaf(-0.5f, gate, c_f);
        ov[k] = fmaf(t, ct, f * cf);
    }
    return ov;
}

__global__ void __launch_bounds__(256)
fuse_kernel(const float* __restrict__ time_out,
            const float* __restrict__ freq_out,
            const float* __restrict__ tfw,   // [2] softmax logits
            const float* __restrict__ gw,    // [2] gate weights
            const float* __restrict__ gb,    // [1] gate bias
            float* __restrict__ out,
            int n4, int rem)
{
    // ---- uniform scalar parameters (scalar pipe: s_load + v_s_exp_f32) ----
    const float a0 = tfw[0], a1 = tfw[1];
    const float m  = fmaxf(a0, a1);
    const float e0 = __expf(a0 - m), e1 = __expf(a1 - m);
    const float w0 = e0 / (e0 + e1);           // softmax weight for time
    const float c_t = 0.5f * w0;               // 0.5*w0
    const float c_f = 1.0f - c_t;              // 0.5*w1 + 0.5  (coef of f before -0.5*gate)
    const float g0 = gw[0], g1 = gw[1], gbv = gb[0];

    const v4f* __restrict__ t4 = (const v4f*)time_out;
    const v4f* __restrict__ f4 = (const v4f*)freq_out;
    v4f* __restrict__ o4 = (v4f*)out;

    const int stride = (int)(gridDim.x * blockDim.x);
    const int gid    = (int)(blockIdx.x * blockDim.x + threadIdx.x);

    // ---- main loop: 2x unrolled grid-stride, 4 b128 loads in flight ----
    int i = gid;
    for (; i + stride < n4; i += 2 * stride) {
        const int j = i + stride;
        // Prefetch next unrolled iteration's 4 chunks (global_prefetch_b8).
        __builtin_prefetch(t4 + i + 2 * stride, 0, 0);
        __builtin_prefetch(f4 + i + 2 * stride, 0, 0);
        __builtin_prefetch(t4 + j + 2 * stride, 0, 0);
        __builtin_prefetch(f4 + j + 2 * stride, 0, 0);

        // 4 independent non-temporal b128 loads before first consume.
        v4f tv0 = __builtin_nontemporal_load(t4 + i);
        v4f fv0 = __builtin_nontemporal_load(f4 + i);
        v4f tv1 = __builtin_nontemporal_load(t4 + j);
        v4f fv1 = __builtin_nontemporal_load(f4 + j);

        v4f ov0 = fuse4(tv0, fv0, g0, g1, gbv, c_t, c_f);
        __builtin_nontemporal_store(ov0, o4 + i);
        v4f ov1 = fuse4(tv1, fv1, g0, g1, gbv, c_t, c_f);
        __builtin_nontemporal_store(ov1, o4 + j);
    }
    // Leftover single chunk for this thread.
    if (i < n4) {
        v4f tv = __builtin_nontemporal_load(t4 + i);
        v4f fv = __builtin_nontemporal_load(f4 + i);
        v4f ov = fuse4(tv, fv, g0, g1, gbv, c_t, c_f);
        __builtin_nontemporal_store(ov, o4 + i);
    }

    // Scalar tail (n % 4 elements; empty for this shape, kept for robustness).
    if (gid < rem) {
        const int idx = n4 * 4 + gid;
        const float t = time_out[idx];
        const float f = freq_out[idx];
        const float z    = fmaf(t, g0, fmaf(f, g1, gbv));
        const float gate = __builtin_amdgcn_rcpf(1.0f + __expf(-z));
        const float ct = fmaf(0.5f, gate, c_t);
        const float cf = fmaf(-0.5f, gate, c_f);
        out[idx] = fmaf(t, ct, f * cf);
    }
}

extern "C" void kernel_launch(void* const* d_in, const int* in_sizes, int n_in,
                              void* d_out, int out_size, void* d_ws, size_t ws_size,
                              hipStream_t stream)
{
    (void)n_in; (void)d_ws; (void)ws_size; (void)in_sizes;
    const float* time_out = (const float*)d_in[0];   // [B,T,N] f32
    const float* freq_out = (const float*)d_in[1];   // [B,T,N] f32
    const float* tfw      = (const float*)d_in[2];   // [2]     f32
    const float* gw       = (const float*)d_in[3];   // [1,2]   f32
    const float* gb       = (const float*)d_in[4];   // [1]     f32
    float*       out      = (float*)d_out;           // [B,T,N] f32

    const int n   = out_size;          // 19,860,480
    const int n4  = n >> 2;            // float4 chunks
    const int rem = n - (n4 << 2);

    const int block = 256;                                   // 8 wave32 per block
    int grid = (n4 + block * 8 - 1) / (block * 8);           // ~8 chunks/thread = 4 unrolled iters
    if (grid < 1) grid = 1;

    fuse_kernel<<<grid, block, 0, stream>>>(time_out, freq_out, tfw, gw, gb, out, n4, rem);
}